// MemoryBank_33818572488863
// MI455X (gfx1250) — compile-verified
//
#include <hip/hip_runtime.h>
#include <hip/hip_bf16.h>

typedef __attribute__((ext_vector_type(2))) float v2f;
typedef __attribute__((ext_vector_type(8))) float v8f;

#define NTOK   8192      // NUM_TOKENS
#define BATCH_ 8192
#define NALL   16384     // NTOK + BATCH
#define TDIM   256
#define NKEYS  3

// ---------------------------------------------------------------------------
// K_sq: sq[i] = |ref_pts[i]|^2
// ---------------------------------------------------------------------------
__global__ __launch_bounds__(256) void k_sq(const float* __restrict__ ref_pts,
                                            float* __restrict__ sq) {
  int i = blockIdx.x * 256 + threadIdx.x;
  if (i < BATCH_) {
    float x = ref_pts[3 * i + 0];
    float y = ref_pts[3 * i + 1];
    float z = ref_pts[3 * i + 2];
    sq[i] = x * x + y * y + z * z;
  }
}

// ---------------------------------------------------------------------------
// K_norm: inv_norm[i] = 1 / max(||token_i||, eps)  over all 16384 rows
// ---------------------------------------------------------------------------
__global__ __launch_bounds__(256) void k_norm(const float* __restrict__ mem_tokens,
                                              const float* __restrict__ new_tokens,
                                              float* __restrict__ inv_norm) {
  int i = blockIdx.x * 256 + threadIdx.x;   // 0..NALL-1
  const float* t = (i < NTOK) ? (mem_tokens + (size_t)i * TDIM)
                              : (new_tokens + (size_t)(i - NTOK) * TDIM);
  const float4* t4 = (const float4*)t;
  float acc = 0.f;
#pragma unroll 4
  for (int k = 0; k < TDIM / 4; ++k) {
    float4 v = t4[k];
    acc += v.x * v.x + v.y * v.y + v.z * v.z + v.w * v.w;
  }
  inv_norm[i] = 1.0f / fmaxf(sqrtf(acc), 1e-12f);
}

// ---------------------------------------------------------------------------
// K_szero / K_saccum: s_vec[d] = sum_i token_i[d] * inv_norm[i]
// ---------------------------------------------------------------------------
__global__ void k_szero(float* __restrict__ s_vec) { s_vec[threadIdx.x] = 0.f; }

__global__ __launch_bounds__(256) void k_saccum(const float* __restrict__ mem_tokens,
                                                const float* __restrict__ new_tokens,
                                                const float* __restrict__ inv_norm,
                                                float* __restrict__ s_vec) {
  int d  = threadIdx.x;              // dim owned by this thread (coalesced)
  int r0 = blockIdx.x * 256;         // 64 blocks x 256 rows
  float acc = 0.f;
  for (int r = 0; r < 256; ++r) {
    int i = r0 + r;
    const float* tok = (i < NTOK) ? (mem_tokens + (size_t)i * TDIM)
                                  : (new_tokens + (size_t)(i - NTOK) * TDIM);
    acc += tok[d] * inv_norm[i];
  }
  atomicAdd(&s_vec[d], acc);
}

// ---------------------------------------------------------------------------
// K_combined: combined[i] = -(token_i . s_vec) * inv_norm[i] + age_i
// (row-sum of the cosine Gram matrix collapsed to a single dot product)
// ---------------------------------------------------------------------------
__global__ __launch_bounds__(256) void k_combined(const float* __restrict__ mem_tokens,
                                                  const float* __restrict__ new_tokens,
                                                  const float* __restrict__ mem_ages,
                                                  const float* __restrict__ inv_norm,
                                                  const float* __restrict__ s_vec,
                                                  float* __restrict__ combined) {
  int i = blockIdx.x * 256 + threadIdx.x;
  const float* t = (i < NTOK) ? (mem_tokens + (size_t)i * TDIM)
                              : (new_tokens + (size_t)(i - NTOK) * TDIM);
  const float4* t4 = (const float4*)t;
  const float4* s4 = (const float4*)s_vec;
  float acc = 0.f;
#pragma unroll 4
  for (int k = 0; k < TDIM / 4; ++k) {
    float4 v = t4[k];
    float4 s = s4[k];
    acc += v.x * s.x + v.y * s.y + v.z * s.z + v.w * s.w;
  }
  float age = (i < NTOK) ? (mem_ages[i] + 1.0f) : 0.0f;
  combined[i] = -(acc * inv_norm[i]) + age;
}

// ---------------------------------------------------------------------------
// K_rank: deterministic top-NTOK-smallest via lexicographic rank.
// rank_i = #{j : c_j < c_i  ||  (c_j == c_i && j < i)}  is a permutation,
// so keep[rank_i] = i reproduces top_k(-combined) ordering incl. ties.
// float4 LDS reads (ds_load_b128): 4 compares per DS op.
// ---------------------------------------------------------------------------
#define RANK_CHUNK 4096
__global__ __launch_bounds__(256) void k_rank(const float* __restrict__ combined,
                                              int* __restrict__ keep) {
  __shared__ float4 sh[RANK_CHUNK / 4];
  int gid = blockIdx.x * 256 + threadIdx.x;    // 0..NALL-1
  float ci = combined[gid];
  int rank = 0;
  for (int base = 0; base < NALL; base += RANK_CHUNK) {
    const float4* src = (const float4*)(combined + base);
    for (int j = threadIdx.x; j < RANK_CHUNK / 4; j += 256) sh[j] = src[j];
    __syncthreads();
    for (int j = 0; j < RANK_CHUNK / 4; ++j) {
      float4 cj = sh[j];
      int jg = base + 4 * j;
      rank += (cj.x < ci || (cj.x == ci && (jg + 0) < gid)) ? 1 : 0;
      rank += (cj.y < ci || (cj.y == ci && (jg + 1) < gid)) ? 1 : 0;
      rank += (cj.z < ci || (cj.z == ci && (jg + 2) < gid)) ? 1 : 0;
      rank += (cj.w < ci || (cj.w == ci && (jg + 3) < gid)) ? 1 : 0;
    }
    __syncthreads();
  }
  if (rank < NTOK) keep[rank] = gid;
}

// ---------------------------------------------------------------------------
// K_gather: out_keys / out_tokens / out_ages = all_*[keep]
// ---------------------------------------------------------------------------
__global__ __launch_bounds__(256) void k_gather(const float* __restrict__ mem_keys,
                                                const float* __restrict__ mem_tokens,
                                                const float* __restrict__ new_tokens,
                                                const float* __restrict__ mem_ages,
                                                const int* __restrict__ keep,
                                                float* __restrict__ out_keys,
                                                float* __restrict__ out_tokens,
                                                float* __restrict__ out_ages) {
  int r = blockIdx.x;          // output row 0..NTOK-1
  int d = threadIdx.x;         // dim 0..255
  int src = keep[r];
  bool from_mem = src < NTOK;
  size_t srow = from_mem ? (size_t)src : (size_t)(src - NTOK);
  const float* tok = from_mem ? (mem_tokens + srow * TDIM) : (new_tokens + srow * TDIM);
  float tv = tok[d];
  out_tokens[(size_t)r * TDIM + d] = tv;
  if (from_mem) {
    const float* kb = mem_keys + srow * NKEYS * TDIM;
    out_keys[((size_t)r * NKEYS + 0) * TDIM + d] = kb[0 * TDIM + d];
    out_keys[((size_t)r * NKEYS + 1) * TDIM + d] = kb[1 * TDIM + d];
    out_keys[((size_t)r * NKEYS + 2) * TDIM + d] = kb[2 * TDIM + d];
  } else {
    // new_keys = broadcast of new_tokens across the key axis
    out_keys[((size_t)r * NKEYS + 0) * TDIM + d] = tv;
    out_keys[((size_t)r * NKEYS + 1) * TDIM + d] = tv;
    out_keys[((size_t)r * NKEYS + 2) * TDIM + d] = tv;
  }
  if (d == 0) out_ages[r] = from_mem ? (mem_ages[src] + 1.0f) : 0.0f;
}

// ---------------------------------------------------------------------------
// K_cdist: top-3 nearest neighbors over 8192 3-D points via
// V_WMMA_F32_16X16X4_F32.
//
// Role swap vs. naive mapping: the block's FIXED 16 points are the N
// dimension, the MOVING tile is M.  C/D layout => every lane's 8 D
// registers all belong to row N = lane&15, columns m = 8*(lane>>4)+v.
// So the per-row top-3 update is pure in-register VALU on all 32 lanes:
// no LDS tile, no barriers, no exec divergence.
//
// K=4 pad slot carries the norms:  A[m][:] = {x,y,z,sq_m},
// B[:][n] = {-2x,-2y,-2z, 1},  C preloaded with sq_n  =>
// D[m][n] = sq_m + sq_n - 2*(p_m.p_n)  straight out of the WMMA.
// sqrt skipped (monotone under max(.,0)).
// ---------------------------------------------------------------------------
__global__ __launch_bounds__(32) void k_cdist_top3(const float* __restrict__ ref_pts,
                                                   const float* __restrict__ sq,
                                                   float* __restrict__ out_closest) {
  const int lane = threadIdx.x;      // 0..31 (wave32)
  const int half = lane >> 4;        // which K-pair / which M-half this lane holds
  const int l16  = lane & 15;
  const int row0 = blockIdx.x * 16;
  const int myrow = row0 + l16;      // fixed point owned by this lane (N dim)

  // B fragment (fixed): B[2h][n], B[2h+1][n]
  v2f b;
  {
    const float* p = ref_pts + (size_t)myrow * 3;
    if (half == 0) { b.x = -2.0f * p[0]; b.y = -2.0f * p[1]; }
    else           { b.x = -2.0f * p[2]; b.y = 1.0f; }
  }
  const float sq_row = sq[myrow];
  const float INF = __builtin_inff();

  float best0 = INF, best1 = INF, best2 = INF;
  int   idx0 = 0, idx1 = 0, idx2 = 0;

  for (int col0 = 0; col0 < BATCH_; col0 += 16) {
    // A fragment (moving): lane loads A[l16][2h], A[l16][2h+1]
    // A[m][:] = {x_m, y_m, z_m, sq_m};  branchless pointer-select, no divergence
    int mpt = col0 + l16;
    const float* p  = ref_pts + (size_t)mpt * 3;
    const float* p1 = half ? (sq + mpt) : (p + 1);
    v2f a;
    a.x = p[half * 2];   // x_m or z_m
    a.y = *p1;           // y_m or sq_m
    v8f c = {sq_row, sq_row, sq_row, sq_row, sq_row, sq_row, sq_row, sq_row};
    c = __builtin_amdgcn_wmma_f32_16x16x4_f32(false, a, false, b,
                                              (short)0, c, false, false);
    // c[v] = d2(row=myrow, col=col0+8*half+v)
#pragma unroll
    for (int v = 0; v < 8; ++v) {
      int col = col0 + 8 * half + v;
      float val = fmaxf(c[v], 0.0f);
      val = (col == myrow) ? INF : val;      // inf diagonal
      // branchless top-3 insert (ascending col order => ties keep low index)
      bool lt0 = val < best0;
      bool lt1 = val < best1;
      bool lt2 = val < best2;
      float nb2 = lt1 ? best1 : (lt2 ? val : best2);
      int   ni2 = lt1 ? idx1  : (lt2 ? col : idx2);
      float nb1 = lt0 ? best0 : (lt1 ? val : best1);
      int   ni1 = lt0 ? idx0  : (lt1 ? col : idx1);
      best0 = lt0 ? val : best0;
      idx0  = lt0 ? col : idx0;
      best1 = nb1; idx1 = ni1;
      best2 = nb2; idx2 = ni2;
    }
  }

  // Merge partner half-wave's top-3 (lane^16 holds the other 8 columns/tile
  // of the SAME row).  Lexicographic (val, idx) compare for exact tie order.
  float pb0 = __shfl_xor(best0, 16); int pi0 = __shfl_xor(idx0, 16);
  float pb1 = __shfl_xor(best1, 16); int pi1 = __shfl_xor(idx1, 16);
  float pb2 = __shfl_xor(best2, 16); int pi2 = __shfl_xor(idx2, 16);
  if (half == 0) {
    float pv[3] = {pb0, pb1, pb2};
    int   pi[3] = {pi0, pi1, pi2};
#pragma unroll
    for (int k = 0; k < 3; ++k) {
      float val = pv[k];
      int   col = pi[k];
      bool lt0 = (val < best0) || (val == best0 && col < idx0);
      bool lt1 = (val < best1) || (val == best1 && col < idx1);
      bool lt2 = (val < best2) || (val == best2 && col < idx2);
      float nb2 = lt1 ? best1 : (lt2 ? val : best2);
      int   ni2 = lt1 ? idx1  : (lt2 ? col : idx2);
      float nb1 = lt0 ? best0 : (lt1 ? val : best1);
      int   ni1 = lt0 ? idx0  : (lt1 ? col : idx1);
      best0 = lt0 ? val : best0;
      idx0  = lt0 ? col : idx0;
      best1 = nb1; idx1 = ni1;
      best2 = nb2; idx2 = ni2;
    }
    out_closest[(size_t)myrow * 3 + 0] = (float)idx0;
    out_closest[(size_t)myrow * 3 + 1] = (float)idx1;
    out_closest[(size_t)myrow * 3 + 2] = (float)idx2;
  }
}

// ---------------------------------------------------------------------------
// kernel_launch
// ---------------------------------------------------------------------------
extern "C" void kernel_launch(void* const* d_in, const int* in_sizes, int n_in,
                              void* d_out, int out_size, void* d_ws, size_t ws_size,
                              hipStream_t stream) {
  (void)in_sizes; (void)n_in; (void)out_size; (void)ws_size;

  const float* new_tokens = (const float*)d_in[0];  // [8192, 256]
  const float* ref_pts    = (const float*)d_in[1];  // [8192, 3]
  const float* mem_keys   = (const float*)d_in[2];  // [8192, 3, 256]
  const float* mem_tokens = (const float*)d_in[3];  // [8192, 256]
  const float* mem_ages   = (const float*)d_in[4];  // [8192]

  // Output layout: keys | tokens | ages | closest_idx (flat, return order)
  float* out_keys    = (float*)d_out;                               // 8192*3*256
  float* out_tokens  = out_keys + (size_t)NTOK * NKEYS * TDIM;      // 8192*256
  float* out_ages    = out_tokens + (size_t)NTOK * TDIM;            // 8192
  float* out_closest = out_ages + NTOK;                             // 8192*3

  // Workspace layout (floats)
  float* ws       = (float*)d_ws;
  float* inv_norm = ws;                    // 16384
  float* s_vec    = ws + NALL;             // 256
  float* combined = ws + NALL + TDIM;      // 16384 (16B aligned)
  int*   keep     = (int*)(ws + 2 * NALL + TDIM);   // 8192 ints
  float* sq       = ws + 2 * NALL + TDIM + NTOK;    // 8192

  // --- cdist path (independent; WMMA) ---
  k_sq<<<BATCH_ / 256, 256, 0, stream>>>(ref_pts, sq);
  k_cdist_top3<<<BATCH_ / 16, 32, 0, stream>>>(ref_pts, sq, out_closest);

  // --- diversity score path (Gram row-sum collapsed to GEMV) ---
  k_norm<<<NALL / 256, 256, 0, stream>>>(mem_tokens, new_tokens, inv_norm);
  k_szero<<<1, TDIM, 0, stream>>>(s_vec);
  k_saccum<<<NALL / 256, 256, 0, stream>>>(mem_tokens, new_tokens, inv_norm, s_vec);
  k_combined<<<NALL / 256, 256, 0, stream>>>(mem_tokens, new_tokens, mem_ages,
                                             inv_norm, s_vec, combined);

  // --- deterministic top-8192 selection + gather ---
  k_rank<<<NALL / 256, 256, 0, stream>>>(combined, keep);
  k_gather<<<NTOK, TDIM, 0, stream>>>(mem_keys, mem_tokens, new_tokens, mem_ages,
                                      keep, out_keys, out_tokens, out_ages);
}